// PRF_61005715472659
// MI455X (gfx1250) — compile-verified
//
#include <hip/hip_runtime.h>

typedef __attribute__((ext_vector_type(2))) float v2f;
typedef __attribute__((ext_vector_type(8))) float v8f;

#define NIMG 256
#define HW   (384 * 384)          // 147456 pixels per image
#define NBIN 256
#define CHUNKS 36                 // 147456 / 36 = 4096 pixels per block
#define PIX_PER_BLOCK (HW / CHUNKS)

// ---------------------------------------------------------------- zero scratch
__global__ void PRF_zero_kernel(unsigned* __restrict__ p, int n) {
    int i = blockIdx.x * blockDim.x + threadIdx.x;
    if (i < n) p[i] = 0u;
}

// ------------------------------------------------- pass 1: per-image histogram
// bin = clip(ceil(pred*255) - 1, -1, 254) + 1 == clip(ceil(pred*255), 0, 255)
__global__ __launch_bounds__(256)
void PRF_hist_kernel(const float* __restrict__ pred,
                     const int*   __restrict__ Y,
                     unsigned*    __restrict__ hist_tp,
                     unsigned*    __restrict__ hist_t1) {
    __shared__ unsigned s_tp[NBIN];
    __shared__ unsigned s_t1[NBIN];
    const int tid = threadIdx.x;
    s_tp[tid] = 0u;
    s_t1[tid] = 0u;
    __syncthreads();

    const int n = blockIdx.y;
    const long base = (long)n * HW + (long)blockIdx.x * PIX_PER_BLOCK;
    const float4* __restrict__ p4 = (const float4*)(pred + base);
    const int4*   __restrict__ y4 = (const int4*)(Y + base);

    // 4096 pixels = 1024 float4 loads; 256 threads x 4 iterations, coalesced.
    #pragma unroll
    for (int it = 0; it < PIX_PER_BLOCK / (256 * 4); ++it) {
        const int idx = it * 256 + tid;
        const float4 p = p4[idx];
        const int4   y = y4[idx];

        int b0 = min(max((int)ceilf(p.x * 255.0f), 0), 255);
        int b1 = min(max((int)ceilf(p.y * 255.0f), 0), 255);
        int b2 = min(max((int)ceilf(p.z * 255.0f), 0), 255);
        int b3 = min(max((int)ceilf(p.w * 255.0f), 0), 255);

        atomicAdd(&s_t1[b0], 1u);
        atomicAdd(&s_t1[b1], 1u);
        atomicAdd(&s_t1[b2], 1u);
        atomicAdd(&s_t1[b3], 1u);
        if (y.x) atomicAdd(&s_tp[b0], 1u);
        if (y.y) atomicAdd(&s_tp[b1], 1u);
        if (y.z) atomicAdd(&s_tp[b2], 1u);
        if (y.w) atomicAdd(&s_tp[b3], 1u);
    }
    __syncthreads();

    const unsigned t1 = s_t1[tid];
    const unsigned tp = s_tp[tid];
    if (t1) atomicAdd(&hist_t1[n * NBIN + tid], t1);
    if (tp) atomicAdd(&hist_tp[n * NBIN + tid], tp);
}

// ---------------------------------------- pass 2: reverse cumsum as WMMA GEMM
// cum[n, j] = sum_b hist[n, b] * (b >= j)   ->  D = A(16x4 tiles) x U + C
// One wave (32 threads) per 16x16 output tile. All counts are integers
// <= 147456, exact in f32, so V_WMMA_F32_16X16X4_F32 is bit-exact here.
__global__ __launch_bounds__(32)
void PRF_cum_wmma_kernel(const unsigned* __restrict__ hist,
                         float*          __restrict__ cum) {
    const unsigned* __restrict__ H = hist + (size_t)blockIdx.z * NIMG * NBIN;
    float*          __restrict__ C = cum  + (size_t)blockIdx.z * NIMG * NBIN;

    const int lane = threadIdx.x;
    const int half = lane >> 4;      // 0: lanes 0-15, 1: lanes 16-31
    const int l    = lane & 15;
    const int m    = blockIdx.y * 16 + l;   // A-matrix row (image) for this lane
    const int nc   = blockIdx.x * 16 + l;   // B/C/D column (bin j) for this lane

    v8f acc = {};
    for (int k0 = 0; k0 < NBIN; k0 += 4) {
        const int ka = k0 + half * 2;       // K pair held by this half-wave
        v2f a, b;
        // A 16x4 f32 layout (ISA 7.12.2): lanes 0-15 hold K=0,1; lanes 16-31 K=2,3
        a.x = (float)H[m * NBIN + ka];
        a.y = (float)H[m * NBIN + ka + 1];
        // B 4x16: U[k, j] = (k >= j); rows split across half-waves like A's K
        b.x = (ka     >= nc) ? 1.0f : 0.0f;
        b.y = (ka + 1 >= nc) ? 1.0f : 0.0f;
        acc = __builtin_amdgcn_wmma_f32_16x16x4_f32(
            /*neg_a=*/false, a, /*neg_b=*/false, b,
            /*c_mod=*/(short)0, acc, /*reuse_a=*/false, /*reuse_b=*/false);
    }

    // C/D 16x16 f32 layout: VGPR r -> row (tile + half*8 + r), col = nc
    const int row0 = blockIdx.y * 16 + half * 8;
    #pragma unroll
    for (int r = 0; r < 8; ++r)
        C[(size_t)(row0 + r) * NBIN + nc] = acc[r];
}

// -------------------------------------------------- pass 3: Ps / Rs / Fs (765)
__global__ __launch_bounds__(256)
void PRF_finalize_kernel(const float* __restrict__ cum_tp,
                         const float* __restrict__ cum_t1,
                         float*       __restrict__ out) {
    const int t = blockIdx.x * blockDim.x + threadIdx.x;
    if (t >= 255) return;
    float ps = 0.0f, rs = 0.0f;
    for (int n = 0; n < NIMG; ++n) {
        const float tp = cum_tp[n * NBIN + (t + 1)];
        const float t1 = cum_t1[n * NBIN + (t + 1)];
        const float t2 = cum_tp[n * NBIN + 0];      // total Y ones = full TP sum
        ps += (t1 > 0.0f) ? tp / t1 : 0.0f;
        rs += (t2 > 0.0f) ? tp / t2 : 0.0f;
    }
    const float Ps = ps * (1.0f / (float)NIMG);
    const float Rs = rs * (1.0f / (float)NIMG);
    const float Fs = 1.3f * Ps * Rs / (Rs + 0.3f * Ps + 1e-9f);
    out[t]       = Ps;
    out[255 + t] = Rs;
    out[510 + t] = Fs;
}

extern "C" void kernel_launch(void* const* d_in, const int* in_sizes, int n_in,
                              void* d_out, int out_size, void* d_ws, size_t ws_size,
                              hipStream_t stream) {
    const float* pred = (const float*)d_in[0];
    const int*   Y    = (const int*)d_in[1];
    float*       out  = (float*)d_out;

    // ws layout (1 MB total):
    //   [0)        hist_tp  u32[256*256]
    //   [256KB)    hist_t1  u32[256*256]
    //   [512KB)    cum_tp   f32[256*256]
    //   [768KB)    cum_t1   f32[256*256]
    unsigned* hist = (unsigned*)d_ws;                      // tp then t1, contiguous
    float*    cum  = (float*)((char*)d_ws + 2u * NIMG * NBIN * sizeof(unsigned));
    unsigned* hist_tp = hist;
    unsigned* hist_t1 = hist + NIMG * NBIN;
    float*    cum_tp  = cum;
    float*    cum_t1  = cum + NIMG * NBIN;

    // 1) zero both histograms (ws is poisoned by the harness)
    {
        const int n = 2 * NIMG * NBIN;                     // 131072 u32
        PRF_zero_kernel<<<(n + 255) / 256, 256, 0, stream>>>(hist, n);
    }

    // 2) streaming histogram: 256 images x 36 chunks, LDS atomics + global merge
    PRF_hist_kernel<<<dim3(CHUNKS, NIMG), 256, 0, stream>>>(pred, Y, hist_tp, hist_t1);

    // 3) reverse cumsum via WMMA: 16x16 col tiles x 16 row tiles x {tp, t1}
    PRF_cum_wmma_kernel<<<dim3(NBIN / 16, NIMG / 16, 2), 32, 0, stream>>>(hist, cum);

    // 4) finalize Ps/Rs/Fs into d_out (765 floats)
    PRF_finalize_kernel<<<1, 256, 0, stream>>>(cum_tp, cum_t1, out);
}